// NuclearChargePairs_37134287242091
// MI455X (gfx1250) — compile-verified
//
#include <hip/hip_runtime.h>
#include <hip/hip_bf16.h>

// NuclearChargePairs: out[b, k] = nc[b, TRI_I[k]] * nc[b, TRI_J[k]]
// B = 512, N_ATOMS = 512, NC2 = 512*511/2 = 130816.
//
// Store-bandwidth-bound (268 MB out @ 23.3 TB/s ~ 11.5 us floor).
// Each wave computes one 16x16 tile of the outer product nc (x) nc via a
// rank-1 V_WMMA_F32_16X16X4_F32 (only K=0 populated -> exact f32 product),
// then streams the strictly-lower-triangular elements out with coalesced,
// non-temporal per-lane b32 stores (output is write-once and > L2 size).
// Off-diagonal tiles (496/528) take a branchless store path; only the 32
// diagonal tiles use the j<i exec mask. Triangular row offsets are formed
// incrementally (f(i+1)-f(i) = i), no per-row multiply.

#define N_ATOMS 512
#define NC2 130816
#define NTILES 528            // 32*33/2 tiles: ti in [0,32), tj <= ti
#define WAVES_PER_BLOCK 8
#define BLOCKS_X (NTILES / WAVES_PER_BLOCK)   // 66

typedef float v2f __attribute__((ext_vector_type(2)));
typedef float v8f __attribute__((ext_vector_type(8)));

__global__ __launch_bounds__(256)
void NuclearChargePairs_37134287242091_kernel(const float* __restrict__ nc,
                                              float* __restrict__ out) {
    __shared__ float row[N_ATOMS];

    const int tid = threadIdx.x;        // 0..255, 8 wave32 waves
    const int b   = blockIdx.y;         // batch

    // Stage the 2 KB batch row into LDS (broadcast source for all 8 tiles).
    const float* __restrict__ nc_b = nc + (size_t)b * N_ATOMS;
    row[tid]       = nc_b[tid];
    row[tid + 256] = nc_b[tid + 256];
    __syncthreads();

    const int lane   = tid & 31;
    const int lane15 = lane & 15;
    const bool lo    = (lane < 16);

    // Tile id is wave-uniform: force it scalar so the triangular-inverse
    // math runs on the SALU.
    const int t = __builtin_amdgcn_readfirstlane(blockIdx.x * WAVES_PER_BLOCK
                                                 + (tid >> 5));   // [0, 528)
    int ti = (int)((sqrtf(8.0f * (float)t + 1.0f) - 1.0f) * 0.5f);
    while ((ti + 1) * (ti + 2) / 2 <= t) ++ti;             // exact fixup
    while (ti * (ti + 1) / 2 > t)        --ti;
    const int tj = t - ti * (ti + 1) / 2;

    const int i0 = ti << 4;   // tile row base   (i dimension)
    const int j0 = tj << 4;   // tile col base   (j dimension)

    // A: 16x4 f32, K=0 slot = (VGPR0, lanes 0-15) -> nc[i0+m]; rest zero.
    // B: 4x16 f32, K=0 slot = (VGPR0, lanes 0-15) -> nc[j0+n]; rest zero.
    // D[m,n] = nc[i0+m]*nc[j0+n] + 0 + 0 + 0  (exact f32 product).
    v2f av, bv;
    av.x = lo ? row[i0 + lane15] : 0.0f;
    av.y = 0.0f;
    bv.x = lo ? row[j0 + lane15] : 0.0f;
    bv.y = 0.0f;
    v8f cv = {0.0f, 0.0f, 0.0f, 0.0f, 0.0f, 0.0f, 0.0f, 0.0f};

    // (neg_a, A, neg_b, B, c_mod, C, reuse_a, reuse_b) — EXEC all-1s here.
    v8f dv = __builtin_amdgcn_wmma_f32_16x16x4_f32(
        false, av, false, bv, (short)0, cv, false, false);

    // D layout: VGPR r holds row M=r (lanes 0-15) and M=r+8 (lanes 16-31),
    // column N = lane&15. Packed triangular offset f(i)+j, f(i)=i*(i-1)/2,
    // built incrementally: f(i+1) = f(i) + i.
    float* __restrict__ out_b = out + (size_t)b * NC2;
    const int j     = j0 + lane15;
    const int rbase = i0 + (lo ? 0 : 8);         // row for r = 0
    int off = (rbase * (rbase - 1)) / 2 + j;     // offset for r = 0

    if (ti != tj) {
        // Fully strictly-lower tile: branchless streaming stores.
#pragma unroll
        for (int r = 0; r < 8; ++r) {
            __builtin_nontemporal_store(dv[r], out_b + off);
            off += rbase + r;                    // advance f(i) -> f(i+1)
        }
    } else {
        // Diagonal tile: keep only j < i.
#pragma unroll
        for (int r = 0; r < 8; ++r) {
            if (j < rbase + r) {
                __builtin_nontemporal_store(dv[r], out_b + off);
            }
            off += rbase + r;
        }
    }
}

extern "C" void kernel_launch(void* const* d_in, const int* in_sizes, int n_in,
                              void* d_out, int out_size, void* d_ws, size_t ws_size,
                              hipStream_t stream) {
    (void)in_sizes; (void)n_in; (void)d_ws; (void)ws_size; (void)out_size;
    const float* nc = (const float*)d_in[0];   // [512, 512] fp32
    float* out      = (float*)d_out;           // [512, 130816] fp32

    dim3 grid(BLOCKS_X, 512, 1);               // 66 blocks x 512 batches
    dim3 block(256, 1, 1);                     // 8 wave32 waves -> 8 tiles
    NuclearChargePairs_37134287242091_kernel<<<grid, block, 0, stream>>>(nc, out);
}